// MultiBoxLoss_51823075393937
// MI455X (gfx1250) — compile-verified
//
#include <hip/hip_runtime.h>
#include <hip/hip_bf16.h>
#include <cmath>

typedef __attribute__((ext_vector_type(2))) float v2f;
typedef __attribute__((ext_vector_type(8))) float v8f;

#define KC 21          // classes (reference C)
#define MAXN 20        // objects per image
#define MAXP 8732      // priors (LDS row buffer in mining kernel)
#define THRESH 0.5f
#define NEGPOS 3
#define VAR0 0.1f
#define VAR1 0.2f

__device__ __forceinline__ float sl1(float x) {
    float ax = fabsf(x);
    return (ax < 1.0f) ? 0.5f * ax * ax : ax - 0.5f;
}

// ---------------- init ----------------
__global__ void k_init(float* accum) {
    if (threadIdx.x == 0) {
        accum[0] = 0.0f;                       // loss_l
        accum[1] = 0.0f;                       // loss_conf
        ((int*)accum)[2] = 0;                  // total_pos
    }
}

// ---------------- kernel 1: matching + loss_l (one block per batch) ----------------
__global__ void k_match(const float* __restrict__ loc,
                        const float* __restrict__ priors,
                        const float* __restrict__ targets,
                        int P, int Nt,
                        int* __restrict__ conf_t,
                        int* __restrict__ num_pos,
                        float* __restrict__ accum) {
    const int b = blockIdx.x;
    const int tid = threadIdx.x;
    __shared__ float s_t[MAXN][4];
    __shared__ int   s_lab[MAXN];
    __shared__ unsigned long long s_key[MAXN];
    __shared__ int   s_bp[MAXN];
    __shared__ float s_red[256];
    __shared__ int   s_redi[256];

    if (tid < Nt && tid < MAXN) {
        const float* tg = targets + ((size_t)b * Nt + tid) * 5;
        s_t[tid][0] = tg[0]; s_t[tid][1] = tg[1];
        s_t[tid][2] = tg[2]; s_t[tid][3] = tg[3];
        s_lab[tid] = (int)tg[4];
        s_key[tid] = 0ull;
    }
    __syncthreads();

    // pass 1: per-truth argmax over priors (first-index tiebreak via ~p)
    unsigned long long bestk[MAXN];
#pragma unroll
    for (int n = 0; n < MAXN; ++n) bestk[n] = 0ull;

    for (int p = tid; p < P; p += 256) {
        const float4 pr = ((const float4*)priors)[p];
        const float px0 = pr.x - 0.5f * pr.z, py0 = pr.y - 0.5f * pr.w;
        const float px1 = pr.x + 0.5f * pr.z, py1 = pr.y + 0.5f * pr.w;
        const float pa = pr.z * pr.w;
#pragma unroll
        for (int n = 0; n < MAXN; ++n) {
            if (n < Nt) {
                float tx0 = s_t[n][0], ty0 = s_t[n][1], tx1 = s_t[n][2], ty1 = s_t[n][3];
                float w = fminf(tx1, px1) - fmaxf(tx0, px0);
                float h = fminf(ty1, py1) - fmaxf(ty0, py0);
                w = fmaxf(w, 0.0f); h = fmaxf(h, 0.0f);
                float inter = w * h;
                float ta = (tx1 - tx0) * (ty1 - ty0);
                float iou = inter / (ta + pa - inter);
                unsigned long long key =
                    ((unsigned long long)__float_as_uint(iou) << 32) |
                    (unsigned int)(~(unsigned int)p);
                if (key > bestk[n]) bestk[n] = key;
            }
        }
    }
#pragma unroll
    for (int n = 0; n < MAXN; ++n)
        if (n < Nt) atomicMax(&s_key[n], bestk[n]);
    __syncthreads();
    if (tid < Nt && tid < MAXN)
        s_bp[tid] = (int)(~(unsigned int)(s_key[tid] & 0xffffffffull));
    __syncthreads();

    // pass 2: per-prior best truth, override with claimed prior, conf + loss_l
    int pc = 0; float ll = 0.0f;
    for (int p = tid; p < P; p += 256) {
        const float4 pr = ((const float4*)priors)[p];
        const float px0 = pr.x - 0.5f * pr.z, py0 = pr.y - 0.5f * pr.w;
        const float px1 = pr.x + 0.5f * pr.z, py1 = pr.y + 0.5f * pr.w;
        const float pa = pr.z * pr.w;
        float bo = -1.0f; int bi = 0;
#pragma unroll
        for (int n = 0; n < MAXN; ++n) {
            if (n < Nt) {
                float tx0 = s_t[n][0], ty0 = s_t[n][1], tx1 = s_t[n][2], ty1 = s_t[n][3];
                float w = fminf(tx1, px1) - fmaxf(tx0, px0);
                float h = fminf(ty1, py1) - fmaxf(ty0, py0);
                w = fmaxf(w, 0.0f); h = fmaxf(h, 0.0f);
                float inter = w * h;
                float ta = (tx1 - tx0) * (ty1 - ty0);
                float iou = inter / (ta + pa - inter);
                if (iou > bo) { bo = iou; bi = n; }   // first-max semantics
            }
        }
        int chosen = -1;
#pragma unroll
        for (int n = 0; n < MAXN; ++n)
            if (n < Nt && s_bp[n] == p) chosen = n;  // max-n semantics
        float bov = bo;
        if (chosen >= 0) { bi = chosen; bov = 2.0f; }
        int conf = (bov < THRESH) ? 0 : (s_lab[bi] + 1);
        conf_t[(size_t)b * P + p] = conf;
        if (conf > 0) {
            ++pc;
            const float4 ld = ((const float4*)loc)[(size_t)b * P + p];
            float tx0 = s_t[bi][0], ty0 = s_t[bi][1], tx1 = s_t[bi][2], ty1 = s_t[bi][3];
            float gcx = ((tx0 + tx1) * 0.5f - pr.x) / (VAR0 * pr.z);
            float gcy = ((ty0 + ty1) * 0.5f - pr.y) / (VAR0 * pr.w);
            float gw = __logf((tx1 - tx0) / pr.z) / VAR1;
            float gh = __logf((ty1 - ty0) / pr.w) / VAR1;
            ll += sl1(ld.x - gcx) + sl1(ld.y - gcy) + sl1(ld.z - gw) + sl1(ld.w - gh);
        }
    }
    s_red[tid] = ll; s_redi[tid] = pc;
    __syncthreads();
    for (int s = 128; s > 0; s >>= 1) {
        if (tid < s) { s_red[tid] += s_red[tid + s]; s_redi[tid] += s_redi[tid + s]; }
        __syncthreads();
    }
    if (tid == 0) {
        num_pos[b] = s_redi[0];
        atomicAdd((int*)accum + 2, s_redi[0]);
        atomicAdd(&accum[0], s_red[0]);
    }
}

// ---------------- kernel 2: streaming logsumexp via WMMA row-sum ----------------
// One wave = 16 priors. A(16x4) holds exp(x-max); B = ones; 6 chained
// V_WMMA_F32_16X16X4_F32 accumulate the 24-padded class sum in f32.
// NO arrays anywhere: pure named-scalar SSA, so SROA can never reintroduce
// dynamic-index compare/select ladders. Each A element is one v_cndmask on h.
__global__ void k_lse(const float* __restrict__ conf,
                      int P,
                      const int* __restrict__ conf_t,
                      float* __restrict__ loss_c,
                      float* __restrict__ ce,
                      const float* __restrict__ accum) {
    const int b = blockIdx.y;
    const int wave = threadIdx.x >> 5;
    const int lane = threadIdx.x & 31;
    const int nTiles = (P + 15) >> 4;
    const int tile = blockIdx.x * 8 + wave;
    if (tile >= nTiles) return;                 // wave-uniform: EXEC stays full

    const bool hb = (lane & 16) != 0;           // K-half for the A layout
    const int m = lane & 15;                    // prior row within tile
    const int pr = tile * 16 + m;
    const int prc = min(pr, P - 1);
    const float* src = conf + ((size_t)b * P + prc) * KC;

    // prefetch one grid-block of tiles ahead (L2 stays ahead of HBM)
    __builtin_prefetch(conf + ((size_t)b * P + min(prc + 16 * 8, P - 1)) * KC, 0, 1);

    const float w0 = src[0],  w1 = src[1],  w2 = src[2],  w3 = src[3];
    const float w4 = src[4],  w5 = src[5],  w6 = src[6],  w7 = src[7];
    const float w8 = src[8],  w9 = src[9],  w10 = src[10], w11 = src[11];
    const float w12 = src[12], w13 = src[13], w14 = src[14], w15 = src[15];
    const float w16 = src[16], w17 = src[17], w18 = src[18], w19 = src[19];
    const float w20 = src[20];

    float mx = fmaxf(fmaxf(fmaxf(fmaxf(w0, w1), fmaxf(w2, w3)),
                           fmaxf(fmaxf(w4, w5), fmaxf(w6, w7))),
                     fmaxf(fmaxf(fmaxf(w8, w9), fmaxf(w10, w11)),
                           fmaxf(fmaxf(w12, w13), fmaxf(w14, w15))));
    mx = fmaxf(mx, fmaxf(fmaxf(fmaxf(w16, w17), fmaxf(w18, w19)), w20));

    const float e0 = __expf(w0 - mx),   e1 = __expf(w1 - mx);
    const float e2 = __expf(w2 - mx),   e3 = __expf(w3 - mx);
    const float e4 = __expf(w4 - mx),   e5 = __expf(w5 - mx);
    const float e6 = __expf(w6 - mx),   e7 = __expf(w7 - mx);
    const float e8 = __expf(w8 - mx),   e9 = __expf(w9 - mx);
    const float e10 = __expf(w10 - mx), e11 = __expf(w11 - mx);
    const float e12 = __expf(w12 - mx), e13 = __expf(w13 - mx);
    const float e14 = __expf(w14 - mx), e15 = __expf(w15 - mx);
    const float e16 = __expf(w16 - mx), e17 = __expf(w17 - mx);
    const float e18 = __expf(w18 - mx), e19 = __expf(w19 - mx);
    const float e20 = __expf(w20 - mx);

    float ls;
#if __has_builtin(__builtin_amdgcn_wmma_f32_16x16x4_f32)
    const v2f ones = {1.0f, 1.0f};
    v8f acc;
    v2f a;
    a.x = hb ? e2 : e0;   a.y = hb ? e3 : e1;
    acc = __builtin_amdgcn_wmma_f32_16x16x4_f32(
        false, a, false, ones, (short)0,
        (v8f){0.f, 0.f, 0.f, 0.f, 0.f, 0.f, 0.f, 0.f}, false, false);
    a.x = hb ? e6 : e4;   a.y = hb ? e7 : e5;
    acc = __builtin_amdgcn_wmma_f32_16x16x4_f32(false, a, false, ones, (short)0, acc, false, false);
    a.x = hb ? e10 : e8;  a.y = hb ? e11 : e9;
    acc = __builtin_amdgcn_wmma_f32_16x16x4_f32(false, a, false, ones, (short)0, acc, false, false);
    a.x = hb ? e14 : e12; a.y = hb ? e15 : e13;
    acc = __builtin_amdgcn_wmma_f32_16x16x4_f32(false, a, false, ones, (short)0, acc, false, false);
    a.x = hb ? e18 : e16; a.y = hb ? e19 : e17;
    acc = __builtin_amdgcn_wmma_f32_16x16x4_f32(false, a, false, ones, (short)0, acc, false, false);
    a.x = hb ? 0.0f : e20; a.y = 0.0f;           // classes 21..23 padded to 0
    acc = __builtin_amdgcn_wmma_f32_16x16x4_f32(false, a, false, ones, (short)0, acc, false, false);

    // D layout: lane0 holds rows 0..7 (col 0), lane16 holds rows 8..15.
    const int srcl = (m < 8) ? 0 : 16;
    const int r = m & 7;
    const float t0 = __shfl(acc.s0, srcl);
    const float t1 = __shfl(acc.s1, srcl);
    const float t2 = __shfl(acc.s2, srcl);
    const float t3 = __shfl(acc.s3, srcl);
    const float t4 = __shfl(acc.s4, srcl);
    const float t5 = __shfl(acc.s5, srcl);
    const float t6 = __shfl(acc.s6, srcl);
    const float t7 = __shfl(acc.s7, srcl);
    const float u0 = (r & 1) ? t1 : t0;
    const float u1 = (r & 1) ? t3 : t2;
    const float u2 = (r & 1) ? t5 : t4;
    const float u3 = (r & 1) ? t7 : t6;
    const float q0 = (r & 2) ? u1 : u0;
    const float q1 = (r & 2) ? u3 : u2;
    ls = (r & 4) ? q1 : q0;
#else
    // fallback: each lane sums its half's classes; pair lanes l, l^16
    float part = (hb ? e2 : e0) + (hb ? e3 : e1) + (hb ? e6 : e4) + (hb ? e7 : e5)
               + (hb ? e10 : e8) + (hb ? e11 : e9) + (hb ? e14 : e12) + (hb ? e15 : e13)
               + (hb ? e18 : e16) + (hb ? e19 : e17) + (hb ? 0.0f : e20);
    ls = part + __shfl_xor(part, 16);
#endif
    const float lse = __logf(ls) + mx;

    if (!hb && pr < P) {
        const size_t flat = (size_t)b * P + pr;
        const int ct = conf_t[flat];
        const float cv = src[ct];               // gather conf at label
        ce[flat] = lse - cv;
        int cl = (ct != 0) ? 1 : 0;
        if (flat == 0 && ((const int*)accum)[2] > 0) cl = 1;
        loss_c[flat] = (ct > 0) ? 0.0f : (lse - ((cl == 1) ? w1 : w0));
    }
}

// ---------------- kernel 3: hard negative mining (one block per batch) ----------------
__global__ void k_mine(const int* __restrict__ conf_t,
                       const float* __restrict__ loss_c,
                       const float* __restrict__ ce,
                       const int* __restrict__ num_pos,
                       int P,
                       float* __restrict__ accum) {
    const int b = blockIdx.x;
    const int tid = threadIdx.x;
    __shared__ float s_lc[MAXP];
    __shared__ int s_cnt[256];
    __shared__ float s_red[256];

    const float* row = loss_c + (size_t)b * P;
    for (int p = tid; p < P; p += 256) s_lc[p] = row[p];
    const int np = num_pos[b];
    const int k = min(NEGPOS * np, P - 1);
    __syncthreads();

    float T = 0.0f; int need_eq = 0;
    if (k > 0) {
        // binary search over float bits for the k-th largest value
        unsigned ans = 0u;
        for (int bit = 30; bit >= 0; --bit) {
            const float fc = __uint_as_float(ans | (1u << bit));
            int c = 0;
            for (int p = tid; p < P; p += 256) c += (s_lc[p] >= fc) ? 1 : 0;
            s_cnt[tid] = c; __syncthreads();
            for (int s = 128; s > 0; s >>= 1) {
                if (tid < s) s_cnt[tid] += s_cnt[tid + s];
                __syncthreads();
            }
            const int tot = s_cnt[0]; __syncthreads();
            if (tot >= k) ans |= (1u << bit);
        }
        T = __uint_as_float(ans);
        int c = 0;
        for (int p = tid; p < P; p += 256) c += (s_lc[p] > T) ? 1 : 0;
        s_cnt[tid] = c; __syncthreads();
        for (int s = 128; s > 0; s >>= 1) {
            if (tid < s) s_cnt[tid] += s_cnt[tid + s];
            __syncthreads();
        }
        need_eq = k - s_cnt[0]; __syncthreads();
    }

    // select: pos  OR  (value > T)  OR  (value == T with smallest indices)
    float sum = 0.0f; int run_eq = 0;
    for (int start = 0; start < P; start += 256) {
        const int p = start + tid;
        const bool inb = p < P;
        const float vv = inb ? s_lc[p] : -1.0f;
        const int flag = (k > 0 && inb && vv == T) ? 1 : 0;
        s_cnt[tid] = flag; __syncthreads();
        for (int off = 1; off < 256; off <<= 1) {
            const int t = (tid >= off) ? s_cnt[tid - off] : 0;
            __syncthreads();
            s_cnt[tid] += t;
            __syncthreads();
        }
        const int excl = s_cnt[tid] - flag;
        const int tot = s_cnt[255];
        __syncthreads();
        const bool sel_eq = flag && (run_eq + excl) < need_eq;
        run_eq += tot;
        const bool pos = inb && conf_t[(size_t)b * P + p] > 0;
        const bool sel = inb && (pos || (k > 0 && (vv > T || sel_eq)));
        if (sel) sum += ce[(size_t)b * P + p];
    }
    s_red[tid] = sum; __syncthreads();
    for (int s = 128; s > 0; s >>= 1) {
        if (tid < s) s_red[tid] += s_red[tid + s];
        __syncthreads();
    }
    if (tid == 0) atomicAdd(&accum[1], s_red[0]);
}

// ---------------- finalize ----------------
__global__ void k_fin(const float* __restrict__ accum, float* __restrict__ out) {
    if (threadIdx.x == 0) {
        const float Nf = (float)((const int*)accum)[2];
        out[0] = accum[0] / Nf;
        out[1] = accum[1] / Nf;
    }
}

extern "C" void kernel_launch(void* const* d_in, const int* in_sizes, int n_in,
                              void* d_out, int out_size, void* d_ws, size_t ws_size,
                              hipStream_t stream) {
    const int P = in_sizes[2] / 4;                       // priors: (P,4)
    const int B = in_sizes[0] / (P * 4);                 // loc: (B,P,4)
    const int Nt = in_sizes[3] / (B * 5);                // targets: (B,N,5)
    const size_t BP = (size_t)B * P;

    const float* loc     = (const float*)d_in[0];
    const float* conf    = (const float*)d_in[1];
    const float* priors  = (const float*)d_in[2];
    const float* targets = (const float*)d_in[3];

    int*   conf_t  = (int*)d_ws;
    float* loss_c  = (float*)d_ws + BP;
    float* ce      = (float*)d_ws + 2 * BP;
    int*   num_pos = (int*)d_ws + 3 * BP;
    float* accum   = (float*)d_ws + 3 * BP + B;

    k_init<<<1, 32, 0, stream>>>(accum);
    k_match<<<B, 256, 0, stream>>>(loc, priors, targets, P, Nt, conf_t, num_pos, accum);

    const int nTiles = (P + 15) >> 4;
    dim3 g2((nTiles + 7) / 8, B);
    k_lse<<<g2, 256, 0, stream>>>(conf, P, conf_t, loss_c, ce, accum);

    k_mine<<<B, 256, 0, stream>>>(conf_t, loss_c, ce, num_pos, P, accum);
    k_fin<<<1, 32, 0, stream>>>(accum, (float*)d_out);
}